// MyGAT_3856880632464
// MI455X (gfx1250) — compile-verified
//
#include <hip/hip_runtime.h>

#define N   23
#define T   12
#define NT  (N * T)        // 276
#define NNT (N * N * T)    // 6348
#define KPAD 24            // K (=m) padded to multiple of 4
#define NPAD 32            // N (=k) padded to two 16-wide tiles
#define ATT_TILE (KPAD * NPAD)

typedef __attribute__((ext_vector_type(2))) float v2f;
typedef __attribute__((ext_vector_type(8))) float v8f;

// ---------------------------------------------------------------------------
// Kernel 1: attention precompute.
// att_t[i][m][k] = attention[i][k][m], zero-padded to 24 (m) x 32 (k).
// One block per i; 23 active lanes each own one (i,k) row over m.
// ---------------------------------------------------------------------------
__global__ void att_precompute_kernel(const float* __restrict__ W,
                                      const int* __restrict__ adj,
                                      float* __restrict__ att_t) {
    const int i   = blockIdx.x;
    const int tid = threadIdx.x;
    float* tile = att_t + (size_t)i * ATT_TILE;

    // Zero the padded tile (covers m==23 row and k>=23 columns).
    for (int idx = tid; idx < ATT_TILE; idx += blockDim.x) tile[idx] = 0.0f;
    __syncthreads();

    if (tid < N) {
        const int k = tid;
        const float* wrow = W + ((size_t)i * N + k) * N;

        // mn = min(0, min_m W[i,k,m])
        float mn = 0.0f;
#pragma unroll
        for (int m = 0; m < N; ++m) mn = fminf(mn, wrow[m]);

        float vals[N];
        float sum = 0.0f;
#pragma unroll
        for (int m = 0; m < N; ++m) {
            // adj with forced self-loops, masking attention[:,k,m]
            int a = (k == m) ? 1 : adj[k * N + m];
            float v = (a != 0) ? (wrow[m] - mn) : 0.0f;
            vals[m] = v;
            sum += v;
        }
        const float inv = 1.0f / sum;
#pragma unroll
        for (int m = 0; m < N; ++m)
            tile[m * NPAD + k] = vals[m] * inv;   // transposed store: [m][k]
    }
}

// ---------------------------------------------------------------------------
// Kernel 2: per-i batched GEMM via V_WMMA_F32_16X16X4_F32.
//   X_i[b, k] = sum_m data[b, m, i] * att_t[i][m][k]
// One block = one 16-batch tile, 24 waves; wave w handles i = w (w < 23).
// A fragment (16x4, M=batch, K=m): lanes 0-15 hold K={0,1}, lanes 16-31 K={2,3}.
// B fragment (4x16, K=m, N=k):     lanes 0-15 hold K={0,1}, lanes 16-31 K={2,3},
//                                  N = lane & 15.
// C/D (16x16 f32): VGPR r -> rows r (lanes 0-15) and r+8 (lanes 16-31).
// ---------------------------------------------------------------------------
__global__ __launch_bounds__(768) void gat_wmma_kernel(
        const float* __restrict__ flow,
        const float* __restrict__ att_t,
        float* __restrict__ out) {
    const int wave = threadIdx.x >> 5;
    if (wave >= N) return;                 // uniform per wave: EXEC stays all-ones
    const int lane = threadIdx.x & 31;

    const int i  = wave;
    const int b0 = blockIdx.x * 16;
    const int Mr = lane & 15;              // row within 16-batch tile / N index for B
    const int kb = (lane >> 4) << 1;       // K sub-offset: 0 (lanes 0-15) or 2 (16-31)

    // lane-private base pointer to data[b0+Mr, 0, i] at the last time step
    const float* base = flow + (size_t)(b0 + Mr) * NNT + i * T + (T - 1);
    const float* att  = att_t + (size_t)i * ATT_TILE;

    v8f c0 = {};   // k-tile 0 (k = 0..15)
    v8f c1 = {};   // k-tile 1 (k = 16..22, zero-padded to 31)

#pragma unroll
    for (int kk = 0; kk < 6; ++kk) {       // K = 23 -> 6 steps of 4
        const int m0 = kk * 4 + kb;
        const int m1 = m0 + 1;

        // A operand: data[b0+Mr, m, i]; clamp+zero the padded m==23 element
        float a0 = base[(m0 < N ? m0 : 0) * NT];
        float a1 = base[(m1 < N ? m1 : 0) * NT];
        a0 = (m0 < N) ? a0 : 0.0f;
        a1 = (m1 < N) ? a1 : 0.0f;
        v2f a = {a0, a1};

        // B operands from the padded, pre-transposed attention tile (rows m>=23 are 0)
        v2f bf0 = { att[m0 * NPAD + Mr],      att[m1 * NPAD + Mr]      };
        v2f bf1 = { att[m0 * NPAD + 16 + Mr], att[m1 * NPAD + 16 + Mr] };

        c0 = __builtin_amdgcn_wmma_f32_16x16x4_f32(false, a, false, bf0,
                                                   (short)0, c0, false, false);
        c1 = __builtin_amdgcn_wmma_f32_16x16x4_f32(false, a, false, bf1,
                                                   (short)0, c1, false, false);
    }

    // Store D: out[b, i, k], out flat = b*N*N + i*N + k
    const int bhi = (lane >> 4) << 3;      // +8 rows for lanes 16-31
#pragma unroll
    for (int r = 0; r < 8; ++r) {
        const int b = b0 + r + bhi;
        float* orow = out + (size_t)b * (N * N) + i * N;
        orow[Mr] = c0[r];
        const int k1 = 16 + Mr;
        if (k1 < N) orow[k1] = c1[r];
    }
}

// ---------------------------------------------------------------------------
extern "C" void kernel_launch(void* const* d_in, const int* in_sizes, int n_in,
                              void* d_out, int out_size, void* d_ws, size_t ws_size,
                              hipStream_t stream) {
    const float* flow = (const float*)d_in[0];   // (B, N, N, T) f32
    const int*   adj  = (const int*)d_in[1];     // (N, N) i32
    const float* W    = (const float*)d_in[2];   // (N, N, N) f32
    float* out   = (float*)d_out;                // (B, N, N, 1) f32
    float* att_t = (float*)d_ws;                 // 23*24*32 floats = 70656 B

    const int B = in_sizes[0] / NNT;             // 16384

    att_precompute_kernel<<<N, 256, 0, stream>>>(W, adj, att_t);
    gat_wmma_kernel<<<B / 16, 768, 0, stream>>>(flow, att_t, out);
}